// MRTAConstraintLoss_76716705841545
// MI455X (gfx1250) — compile-verified
//
#include <hip/hip_runtime.h>
#include <cstdint>

typedef float v2f __attribute__((ext_vector_type(2)));
typedef float v4f __attribute__((ext_vector_type(4)));
typedef float v8f __attribute__((ext_vector_type(8)));

#define MAIN_BLOCKS 1024
#define MAIN_THREADS 256

// numerically-stable softplus(x) - x*t
__device__ __forceinline__ float bce_term(float x, float t) {
    float sp = fmaxf(x, 0.0f) + __logf(1.0f + __expf(-fabsf(x)));
    return sp - x * t;
}

__global__ __launch_bounds__(MAIN_THREADS) void mrta_main_kernel(
    const float* __restrict__ pred, const float* __restrict__ tgt,
    const float* __restrict__ R, const float* __restrict__ Q,
    float* __restrict__ blk_base, float* __restrict__ blk_viol, int B)
{
    const int tid = blockIdx.x * blockDim.x + threadIdx.x;
    const int nthreads = gridDim.x * blockDim.x;

    float base_acc = 0.0f;   // sum of softplus(p)-p*t over this thread's elements
    float viol_acc = 0.0f;   // sum of masked shortages

    for (long i = tid; i < B; i += nthreads) {
        // pred/tgt/R rows are 96B (16B-aligned) -> 6x b128 non-temporal loads each
        const v4f* p4 = (const v4f*)(pred + i * 24);
        const v4f* t4 = (const v4f*)(tgt + i * 24);
        const v4f* r4 = (const v4f*)(R + i * 24);
        float pv[24], tv[24], rv[24];
#pragma unroll
        for (int k = 0; k < 6; ++k) {
            v4f a = __builtin_nontemporal_load(p4 + k);
            v4f b = __builtin_nontemporal_load(t4 + k);
            v4f c = __builtin_nontemporal_load(r4 + k);
            pv[4 * k + 0] = a.x; pv[4 * k + 1] = a.y; pv[4 * k + 2] = a.z; pv[4 * k + 3] = a.w;
            tv[4 * k + 0] = b.x; tv[4 * k + 1] = b.y; tv[4 * k + 2] = b.z; tv[4 * k + 3] = b.w;
            rv[4 * k + 0] = c.x; rv[4 * k + 1] = c.y; rv[4 * k + 2] = c.z; rv[4 * k + 3] = c.w;
        }
        // Q row is 36B (only 4B aligned) -> scalar loads
        const float* q = Q + i * 9;
        float q9[9];
#pragma unroll
        for (int k = 0; k < 9; ++k) q9[k] = __builtin_nontemporal_load(q + k);

        // BCE-with-logits partial
#pragma unroll
        for (int k = 0; k < 24; ++k) base_acc += bce_term(pv[k], tv[k]);

        // capability violation: assigned(8x3) @ Q(3x3), hinge vs R, active mask
#pragma unroll
        for (int tt = 0; tt < 8; ++tt) {
            float p0 = pv[tt * 3 + 0], p1 = pv[tt * 3 + 1], p2 = pv[tt * 3 + 2];
            float rowsum = p0 + p1 + p2;
            bool a0 = p0 > 0.5f, a1 = p1 > 0.5f, a2 = p2 > 0.5f;
            float sh = 0.0f;
#pragma unroll
            for (int c = 0; c < 3; ++c) {
                float cap = (a0 ? q9[0 * 3 + c] : 0.0f)
                          + (a1 ? q9[1 * 3 + c] : 0.0f)
                          + (a2 ? q9[2 * 3 + c] : 0.0f);
                sh += fmaxf(rv[tt * 3 + c] - cap, 0.0f);
            }
            viol_acc += (rowsum > 0.0f) ? sh : 0.0f;
        }
    }

    // ---- wave-level reduction via V_WMMA_F32_16X16X4_F32 ----
    // A (16x4 f32, 2 VGPRs): VGPR0 -> A[m][0]=base[lane m], A[m][2]=base[lane m+16]
    //                        VGPR1 -> A[m][1]=viol[lane m], A[m][3]=viol[lane m+16]
    // B (4x16 f32): ones at (k=0,n=0),(k=2,n=0) and (k=1,n=1),(k=3,n=1)
    //   -> C[m][0] = base[m]+base[m+16], C[m][1] = viol[m]+viol[m+16]
    const int lane = threadIdx.x & 31;
    const int wave = threadIdx.x >> 5;

    v2f a;
    a.x = base_acc;
    a.y = viol_acc;
    v2f b;
    b.x = (lane == 0 || lane == 16) ? 1.0f : 0.0f;  // B[0][0], B[2][0]
    b.y = (lane == 1 || lane == 17) ? 1.0f : 0.0f;  // B[1][1], B[3][1]
    v8f c = {};
    c = __builtin_amdgcn_wmma_f32_16x16x4_f32(
        /*neg_a=*/false, a, /*neg_b=*/false, b,
        /*c_mod=*/(short)0, c, /*reuse_a=*/false, /*reuse_b=*/false);

    // lane 0: C[0..7][0], lane 16: C[8..15][0] (base); lanes 1/17 same for viol
    float s = c[0] + c[1] + c[2] + c[3] + c[4] + c[5] + c[6] + c[7];

    __shared__ float s_base[16];
    __shared__ float s_viol[16];
    if (lane == 0)  s_base[wave * 2 + 0] = s;
    if (lane == 16) s_base[wave * 2 + 1] = s;
    if (lane == 1)  s_viol[wave * 2 + 0] = s;
    if (lane == 17) s_viol[wave * 2 + 1] = s;
    __syncthreads();

    if (threadIdx.x == 0) {
        float bs = 0.0f, vs = 0.0f;
#pragma unroll
        for (int k = 0; k < 16; ++k) { bs += s_base[k]; vs += s_viol[k]; }
        blk_base[blockIdx.x] = bs;
        blk_viol[blockIdx.x] = vs;
    }
}

__global__ __launch_bounds__(256) void mrta_finalize_kernel(
    const float* __restrict__ blk_base, const float* __restrict__ blk_viol,
    float* __restrict__ out, int nblocks, int B)
{
    __shared__ double sb[256];
    __shared__ double sv[256];
    double bsum = 0.0, vsum = 0.0;
    for (int i = threadIdx.x; i < nblocks; i += 256) {
        bsum += (double)blk_base[i];
        vsum += (double)blk_viol[i];
    }
    sb[threadIdx.x] = bsum;
    sv[threadIdx.x] = vsum;
    __syncthreads();
    for (int off = 128; off > 0; off >>= 1) {
        if (threadIdx.x < off) {
            sb[threadIdx.x] += sb[threadIdx.x + off];
            sv[threadIdx.x] += sv[threadIdx.x + off];
        }
        __syncthreads();
    }
    if (threadIdx.x == 0) {
        double base_loss = sb[0] / ((double)B * 24.0);
        double cap_viol  = sv[0] / (double)B;
        double total     = base_loss + 0.1 * cap_viol;
        out[0] = (float)total;
        out[1] = (float)base_loss;
        out[2] = (float)cap_viol;
        out[3] = 0.0f;  // time_violation
    }
}

extern "C" void kernel_launch(void* const* d_in, const int* in_sizes, int n_in,
                              void* d_out, int out_size, void* d_ws, size_t ws_size,
                              hipStream_t stream) {
    const float* pred = (const float*)d_in[0];  // [B, 24]
    const float* tgt  = (const float*)d_in[1];  // [B, 24]
    const float* R    = (const float*)d_in[2];  // [B, 8, 3]
    const float* Q    = (const float*)d_in[3];  // [B, 3, 3]
    const int B = in_sizes[0] / 24;

    float* blk_base = (float*)d_ws;             // [MAIN_BLOCKS]
    float* blk_viol = blk_base + MAIN_BLOCKS;   // [MAIN_BLOCKS]

    mrta_main_kernel<<<MAIN_BLOCKS, MAIN_THREADS, 0, stream>>>(
        pred, tgt, R, Q, blk_base, blk_viol, B);
    mrta_finalize_kernel<<<1, 256, 0, stream>>>(
        blk_base, blk_viol, (float*)d_out, MAIN_BLOCKS, B);
}